// Pooler_91285234909776
// MI455X (gfx1250) — compile-verified
//
#include <hip/hip_runtime.h>
#include <hip/hip_bf16.h>

typedef float v4f __attribute__((ext_vector_type(4)));
typedef int   v4i __attribute__((ext_vector_type(4)));

#define AS1 __attribute__((address_space(1)))
#define AS3 __attribute__((address_space(3)))

// ---------------------------------------------------------------------------
// gfx1250 async LDS->global store (ASYNCcnt path). Builtin signature (from
// clang diagnostics): (global v4i*, local v4i*, imm offset, imm cpol).
// ---------------------------------------------------------------------------
__device__ __forceinline__ void async_store_b128(float* gaddr, void* lds_src) {
#if __has_builtin(__builtin_amdgcn_global_store_async_from_lds_b128)
    __builtin_amdgcn_global_store_async_from_lds_b128(
        (AS1 v4i*)(AS1 void*)gaddr,
        (AS3 v4i*)(AS3 void*)lds_src, 0, 0);
#else
    unsigned lds_off = (unsigned)(__UINTPTR_TYPE__)(AS3 void*)lds_src;
    asm volatile("global_store_async_from_lds_b128 %0, %1, off"
                 :: "v"(gaddr), "v"(lds_off)
                 : "memory");
#endif
}

__device__ __forceinline__ void wait_asynccnt0() {
#if __has_builtin(__builtin_amdgcn_s_wait_asynccnt)
    __builtin_amdgcn_s_wait_asynccnt(0);
#else
    asm volatile("s_wait_asynccnt 0x0" ::: "memory");
#endif
}

__device__ __forceinline__ v4f vmax4(v4f a, v4f b) {
    v4f r;
    r.x = fmaxf(a.x, b.x);
    r.y = fmaxf(a.y, b.y);
    r.z = fmaxf(a.z, b.z);
    r.w = fmaxf(a.w, b.w);
    return r;
}

// ---------------------------------------------------------------------------
// Kernel 1: exclusive prefix sum of 4096 segment lengths (one block).
// Each of 1024 threads owns 4 lengths; Hillis-Steele scan of the 1024
// per-thread partial sums in LDS, then write out 4 exclusive offsets.
// ---------------------------------------------------------------------------
__global__ __launch_bounds__(1024) void scan_kernel(const int* __restrict__ lengths,
                                                    int* __restrict__ offsets) {
    __shared__ int sums[1024];
    const int t = threadIdx.x;
    const v4i l = ((const v4i*)lengths)[t];
    sums[t] = l.x + l.y + l.z + l.w;
    __syncthreads();
    for (int off = 1; off < 1024; off <<= 1) {
        int v = (t >= off) ? sums[t - off] : 0;
        __syncthreads();
        sums[t] += v;
        __syncthreads();
    }
    const int excl = (t == 0) ? 0 : sums[t - 1];
    offsets[4 * t + 0] = excl;
    offsets[4 * t + 1] = excl + l.x;
    offsets[4 * t + 2] = excl + l.x + l.y;
    offsets[4 * t + 3] = excl + l.x + l.y + l.z;
}

// ---------------------------------------------------------------------------
// Kernel 2: one workgroup (8 waves) per segment.
// Phase 1: each wave strides over rows; lane owns columns [4*lane, 4*lane+4)
//          -> 512B contiguous per wave per row, non-temporal loads.
// Phase 2: cross-wave max reduction through LDS into a pooled row in LDS.
// Phase 3: broadcast pooled row to all `len` output rows via async
//          LDS->global B128 stores (addresses only, data never in VGPRs).
// ---------------------------------------------------------------------------
__global__ __launch_bounds__(256) void pool_kernel(const v4f* __restrict__ h4,
                                                   const int* __restrict__ lengths,
                                                   const int* __restrict__ offsets,
                                                   float* __restrict__ out) {
    __shared__ v4f partial[8][32];
    __shared__ v4f pooled[32];

    const int seg  = blockIdx.x;
    const int lane = threadIdx.x & 31;
    const int wave = threadIdx.x >> 5;
    const int start = offsets[seg];
    const int len   = lengths[seg];

    // Phase 1: strided ragged max, 2-deep unroll for load ILP.
    const v4f* base = h4 + (size_t)start * 32 + lane;
    const float ninf = -__builtin_inff();
    v4f acc0 = {ninf, ninf, ninf, ninf};
    v4f acc1 = acc0;
    int r = wave;
    for (; r + 8 < len; r += 16) {
        v4f a = __builtin_nontemporal_load(base + (size_t)r * 32);
        v4f b = __builtin_nontemporal_load(base + (size_t)(r + 8) * 32);
        acc0 = vmax4(acc0, a);
        acc1 = vmax4(acc1, b);
    }
    if (r < len) {
        v4f a = __builtin_nontemporal_load(base + (size_t)r * 32);
        acc0 = vmax4(acc0, a);
    }
    acc0 = vmax4(acc0, acc1);

    // Phase 2: reduce the 8 per-wave partials into the pooled LDS row.
    partial[wave][lane] = acc0;
    __syncthreads();
    if (threadIdx.x < 32) {
        v4f m = partial[0][lane];
#pragma unroll
        for (int w = 1; w < 8; ++w)
            m = vmax4(m, partial[w][lane]);
        pooled[lane] = m;
    }
    __syncthreads();

    // Phase 3: repeat_interleave via async LDS->global stores.
    float* orow = out + (size_t)start * 128 + lane * 4;
    void* lsrc = (void*)&pooled[lane];
    for (int rr = wave; rr < len; rr += 8) {
        async_store_b128(orow + (size_t)rr * 128, lsrc);
    }
    wait_asynccnt0();
}

extern "C" void kernel_launch(void* const* d_in, const int* in_sizes, int n_in,
                              void* d_out, int out_size, void* d_ws, size_t ws_size,
                              hipStream_t stream) {
    const float* h       = (const float*)d_in[0];
    const int*   lengths = (const int*)d_in[1];
    float*       out     = (float*)d_out;
    int*         offsets = (int*)d_ws;          // 4096 ints of scratch
    const int    n_seg   = in_sizes[1];         // 4096

    scan_kernel<<<1, 1024, 0, stream>>>(lengths, offsets);
    pool_kernel<<<n_seg, 256, 0, stream>>>((const v4f*)h, lengths, offsets, out);
}